// Point_Branch_TF_16106127360572
// MI455X (gfx1250) — compile-verified
//
#include <hip/hip_runtime.h>
#include <hip/hip_bf16.h>

// ---------------------------------------------------------------------------
// CDNA5 (gfx1250) implementation of the Point-Transformer branch.
// fp32 WMMA (v_wmma_f32_16x16x4_f32) used for all GEMM-like stages.
// ---------------------------------------------------------------------------

#define BB    8          // batch
#define NN    8192       // points
#define CI    64         // in channels
#define CO    128        // out channels
#define MM    1024       // sampled centers
#define KK    32         // neighbors
#define HH    8          // heads
#define DD    16         // head dim

typedef __attribute__((ext_vector_type(2))) float v2f;
typedef __attribute__((ext_vector_type(8))) float v8f;

static __device__ __forceinline__ v8f vzero8() {
  v8f z = {0.f, 0.f, 0.f, 0.f, 0.f, 0.f, 0.f, 0.f};
  return z;
}

// D = A(16x4,f32) x B(4x16,f32) + C
static __device__ __forceinline__ v8f wmma4(v2f a, v2f b, v8f c) {
  return __builtin_amdgcn_wmma_f32_16x16x4_f32(
      /*neg_a=*/false, a, /*neg_b=*/false, b,
      /*c_mod=*/(short)0, c, /*reuse_a=*/false, /*reuse_b=*/false);
}

// ---------------------------------------------------------------------------
// 1) Farthest-point sampling: one block per batch, sequential argmax scan.
// ---------------------------------------------------------------------------
__global__ void fps_kernel(const float* __restrict__ coords, int* __restrict__ idx) {
  int b = blockIdx.x;
  int tid = threadIdx.x;
  __shared__ float dist[NN];
  __shared__ float rv[256];
  __shared__ int ri[256];
  __shared__ float lp[3];
  __shared__ int lastSh;

  const float* X = coords + (size_t)(b * 3 + 0) * NN;
  const float* Y = coords + (size_t)(b * 3 + 1) * NN;
  const float* Z = coords + (size_t)(b * 3 + 2) * NN;

  for (int n = tid; n < NN; n += 256) dist[n] = 1e10f;
  if (tid == 0) { idx[b * MM] = 0; lastSh = 0; }
  __syncthreads();

  for (int s = 1; s < MM; s++) {
    if (tid == 0) {
      int last = lastSh;
      lp[0] = X[last]; lp[1] = Y[last]; lp[2] = Z[last];
    }
    __syncthreads();
    float lx = lp[0], ly = lp[1], lz = lp[2];
    float bv = -1.f; int bi = 0;
    for (int n = tid; n < NN; n += 256) {
      float dx = X[n] - lx, dy = Y[n] - ly, dz = Z[n] - lz;
      float d = dx * dx + dy * dy + dz * dz;
      float nd = fminf(dist[n], d);
      dist[n] = nd;
      if (nd > bv) { bv = nd; bi = n; }
    }
    rv[tid] = bv; ri[tid] = bi;
    __syncthreads();
    for (int st = 128; st > 0; st >>= 1) {
      if (tid < st) {
        float v2 = rv[tid + st]; int j2 = ri[tid + st];
        if (v2 > rv[tid] || (v2 == rv[tid] && j2 < ri[tid])) { rv[tid] = v2; ri[tid] = j2; }
      }
      __syncthreads();
    }
    if (tid == 0) { lastSh = ri[0]; idx[b * MM + s] = ri[0]; }
    __syncthreads();
  }
}

// ---------------------------------------------------------------------------
// 2) Gather center coords (b,3,m) and center features as rows (b*m, 64).
// ---------------------------------------------------------------------------
__global__ void gather_centers(const float* __restrict__ coords,
                               const float* __restrict__ feats,
                               const int* __restrict__ fidx,
                               float* __restrict__ cc, float* __restrict__ cfR) {
  int t = blockIdx.x * 256 + threadIdx.x;   // 0 .. B*M-1
  int b = t >> 10, mi = t & (MM - 1);
  int id = fidx[t];
  for (int d = 0; d < 3; d++)
    cc[(b * 3 + d) * MM + mi] = coords[(size_t)(b * 3 + d) * NN + id];
  for (int c = 0; c < CI; c++)
    cfR[(size_t)t * CI + c] = feats[(size_t)(b * CI + c) * NN + id];
}

// ---------------------------------------------------------------------------
// 3) Ball query: first K=32 indices (increasing) with d2 < r^2.
// ---------------------------------------------------------------------------
__global__ void ball_query(const float* __restrict__ coords,
                           const float* __restrict__ cc, int* __restrict__ nbr) {
  int t = blockIdx.x * 256 + threadIdx.x;
  int b = t >> 10, mi = t & (MM - 1);
  float cx = cc[(b * 3 + 0) * MM + mi];
  float cy = cc[(b * 3 + 1) * MM + mi];
  float cz = cc[(b * 3 + 2) * MM + mi];
  const float* X = coords + (size_t)(b * 3 + 0) * NN;
  const float* Y = coords + (size_t)(b * 3 + 1) * NN;
  const float* Z = coords + (size_t)(b * 3 + 2) * NN;
  int buf[KK];
  int cnt = 0;
  for (int n = 0; n < NN && cnt < KK; n++) {
    float dx = X[n] - cx, dy = Y[n] - cy, dz = Z[n] - cz;
    float d = dx * dx + dy * dy + dz * dz;
    if (d < 0.04f) buf[cnt++] = n;
  }
  if (cnt == 0) {
    for (int k = 0; k < KK; k++) nbr[t * KK + k] = NN - 1;
  } else {
    int f = buf[0];
    for (int k = 0; k < KK; k++) nbr[t * KK + k] = (k < cnt) ? buf[k] : f;
  }
}

// ---------------------------------------------------------------------------
// 4) Fused local attention per center (one block per (b,m)):
//    edge build -> q/k/v WMMA -> energy WMMA -> softmax/renorm ->
//    v^T*attn WMMA -> Wtc WMMA -> store x_r + BN2 partial sums.
// ---------------------------------------------------------------------------
__global__ void __launch_bounds__(256)
local_attn(const float* __restrict__ feats, const float* __restrict__ cfR,
           const int* __restrict__ nbr,
           const float* __restrict__ Wq, const float* __restrict__ bq,
           const float* __restrict__ Wk, const float* __restrict__ bk,
           const float* __restrict__ Wv, const float* __restrict__ bv,
           const float* __restrict__ Wtc, const float* __restrict__ btc,
           float* __restrict__ xrOut, float* __restrict__ psum, float* __restrict__ psq) {
  alignas(16) __shared__ float e[KK * 2 * CI];   // 32x128, reused as y[128][32]
  alignas(16) __shared__ float qs[KK * CO];      // 32x128
  alignas(16) __shared__ float ks[KK * CO];
  alignas(16) __shared__ float vs[KK * CO];
  alignas(16) __shared__ float at[KK * KK];      // 32x32
  alignas(16) __shared__ float x1[CO * KK];      // 128x32
  __shared__ float cfl[CI];
  __shared__ int nb[KK];

  int bm = blockIdx.x;
  int b = bm >> 10;
  int tid = threadIdx.x;
  int wid = tid >> 5, lane = tid & 31, half = lane >> 4, ln = lane & 15;

  if (tid < KK) nb[tid] = nbr[bm * KK + tid];
  if (tid >= 64 && tid < 128) cfl[tid - 64] = cfR[(size_t)bm * CI + (tid - 64)];
  __syncthreads();

  // edge rows: e[k][c] = c<64 ? feat[b][c][nb[k]] - cf[c] : cf[c-64]
  for (int t = tid; t < KK * 128; t += 256) {
    int kkk = t >> 7, c = t & 127;
    int c6 = c & 63;
    float fv = feats[(size_t)(b * CI + c6) * NN + nb[kkk]];
    float x = cfl[c6];
    e[t] = (c < 64) ? (fv - x) : x;
  }
  __syncthreads();

  // q,k,v: 48 tiles (3 mats * 2 Mtiles * 8 Ntiles), 6 per wave
  for (int i = 0; i < 6; i++) {
    int T = wid * 6 + i;
    int mat = T >> 4, tt = T & 15, mt = tt >> 3, nt = tt & 7;
    const float* Wm = (mat == 0) ? Wq : ((mat == 1) ? Wk : Wv);
    const float* bm_ = (mat == 0) ? bq : ((mat == 1) ? bk : bv);
    float* outp = (mat == 0) ? qs : ((mat == 1) ? ks : vs);
    const float* Al = e + (mt * 16 + ln) * 128 + 2 * half;
    const float* Wl = Wm + (nt * 16 + ln) * 128 + 2 * half;
    v8f acc = vzero8();
    for (int k0 = 0; k0 < 128; k0 += 4) {
      v2f a = *(const v2f*)(Al + k0);
      v2f bf = *(const v2f*)(Wl + k0);
      acc = wmma4(a, bf, acc);
    }
    float bvv = bm_[nt * 16 + ln];
    for (int r = 0; r < 8; r++)
      outp[(mt * 16 + r + 8 * half) * 128 + nt * 16 + ln] = acc[r] + bvv;
  }
  __syncthreads();

  // energy[kq][j] = q . k  (32x32), 4 tiles on waves 0..3
  if (wid < 4) {
    int mt = wid >> 1, nt = wid & 1;
    const float* Al = qs + (mt * 16 + ln) * 128 + 2 * half;
    const float* Bl = ks + (nt * 16 + ln) * 128 + 2 * half;
    v8f acc = vzero8();
    for (int k0 = 0; k0 < 128; k0 += 4) {
      v2f a = *(const v2f*)(Al + k0);
      v2f bf = *(const v2f*)(Bl + k0);
      acc = wmma4(a, bf, acc);
    }
    for (int r = 0; r < 8; r++)
      at[(mt * 16 + r + 8 * half) * KK + nt * 16 + ln] = acc[r];
  }
  __syncthreads();

  // row softmax (over j), then column renorm (over k)
  if (tid < KK) {
    float mx = -3e38f;
    for (int j = 0; j < KK; j++) mx = fmaxf(mx, at[tid * KK + j]);
    float s = 0.f;
    for (int j = 0; j < KK; j++) { float p = __expf(at[tid * KK + j] - mx); at[tid * KK + j] = p; s += p; }
    float inv = 1.f / s;
    for (int j = 0; j < KK; j++) at[tid * KK + j] *= inv;
  }
  __syncthreads();
  if (tid < KK) {
    float s = 1e-9f;
    for (int k = 0; k < KK; k++) s += at[k * KK + tid];
    float inv = 1.f / s;
    for (int k = 0; k < KK; k++) at[k * KK + tid] *= inv;
  }
  __syncthreads();

  // x1[c][j] = sum_k v[k][c] * attn[k][j]   (128x32, K=32) : 16 tiles, 2/wave
  for (int i = 0; i < 2; i++) {
    int T = wid * 2 + i, mt = T >> 1, nt = T & 1;
    v8f acc = vzero8();
    for (int k0 = 0; k0 < KK; k0 += 4) {
      v2f a, bf;
      a.x = vs[(k0 + 2 * half) * 128 + mt * 16 + ln];
      a.y = vs[(k0 + 2 * half + 1) * 128 + mt * 16 + ln];
      bf.x = at[(k0 + 2 * half) * KK + nt * 16 + ln];
      bf.y = at[(k0 + 2 * half + 1) * KK + nt * 16 + ln];
      acc = wmma4(a, bf, acc);
    }
    for (int r = 0; r < 8; r++)
      x1[(mt * 16 + r + 8 * half) * KK + nt * 16 + ln] = acc[r];
  }
  __syncthreads();

  // y[o][j] = Wtc[o][:] . x1[:][j] + btc[o]  (128x32, K=128) : 16 tiles, 2/wave
  for (int i = 0; i < 2; i++) {
    int T = wid * 2 + i, mt = T >> 1, nt = T & 1;
    const float* Wl = Wtc + (mt * 16 + ln) * 128 + 2 * half;
    v8f acc = vzero8();
    for (int k0 = 0; k0 < 128; k0 += 4) {
      v2f a = *(const v2f*)(Wl + k0);
      v2f bf;
      bf.x = x1[(k0 + 2 * half) * KK + nt * 16 + ln];
      bf.y = x1[(k0 + 2 * half + 1) * KK + nt * 16 + ln];
      acc = wmma4(a, bf, acc);
    }
    for (int r = 0; r < 8; r++) {
      int o = mt * 16 + r + 8 * half;
      int j = nt * 16 + ln;
      float val = acc[r] + btc[o];
      xrOut[(size_t)bm * (CO * KK) + o * KK + j] = val;
      e[o * KK + j] = val;  // reuse e as y[128][32]
    }
  }
  __syncthreads();

  if (tid < CO) {
    float s = 0.f, q = 0.f;
    for (int j = 0; j < KK; j++) { float v = e[tid * KK + j]; s += v; q += v * v; }
    psum[(size_t)bm * CO + tid] = s;
    psq[(size_t)bm * CO + tid] = q;
  }
}

// ---------------------------------------------------------------------------
// 5) Deterministic column reduction (128 cols). mode 0: sum; mode 1: sum+sumsq
// ---------------------------------------------------------------------------
__global__ void col_reduce(const float* __restrict__ X, int rows,
                           float* __restrict__ outSum, float* __restrict__ outSq,
                           int withSq) {
  int c = blockIdx.x;
  int t = threadIdx.x;
  __shared__ float sh[256], shq[256];
  float s = 0.f, q = 0.f;
  for (int r = t; r < rows; r += 256) {
    float v = X[(size_t)r * 128 + c];
    s += v;
    if (withSq) q += v * v;
  }
  sh[t] = s; shq[t] = q;
  __syncthreads();
  for (int st = 128; st > 0; st >>= 1) {
    if (t < st) { sh[t] += sh[t + st]; shq[t] += shq[t + st]; }
    __syncthreads();
  }
  if (t == 0) { outSum[c] = sh[0]; if (withSq) outSq[c] = shq[0]; }
}

// ---------------------------------------------------------------------------
// 6) BN2 finalize + relu + max/mean pool over K -> xm_in rows (b*m, 256)
// ---------------------------------------------------------------------------
__global__ void bn2_pool(const float* __restrict__ xr, const float* __restrict__ sum,
                         const float* __restrict__ sq, const float* __restrict__ g,
                         const float* __restrict__ be, float* __restrict__ xmIn) {
  int bm = blockIdx.x;
  int c = threadIdx.x;  // 128 threads
  const float invc = 1.f / 262144.f;
  float mu = sum[c] * invc;
  float var = sq[c] * invc - mu * mu;
  float sc = rsqrtf(var + 1e-5f) * g[c];
  float sh = be[c] - mu * sc;
  const float* p = xr + (size_t)bm * (CO * KK) + c * KK;
  float mx = -3e38f, sm = 0.f;
  for (int j = 0; j < KK; j++) {
    float v = fmaxf(p[j] * sc + sh, 0.f);
    mx = fmaxf(mx, v);
    sm += v;
  }
  xmIn[(size_t)bm * 256 + c] = mx;
  xmIn[(size_t)bm * 256 + 128 + c] = sm * (1.f / 32.f);
}

// ---------------------------------------------------------------------------
// 7) Generic WMMA GEMM: Out(row,128) = A(row,K) @ W(128,K)^T + bias
//    mode 0: row-major out; mode 1: head-split [b][h][m][16]
// ---------------------------------------------------------------------------
__global__ void __launch_bounds__(256)
gemm_wmma(const float* __restrict__ A, int lda, const float* __restrict__ W, int ldw,
          const float* __restrict__ bias, float* __restrict__ Out, int Kdim, int mode) {
  int tid = threadIdx.x;
  int wid = tid >> 5, lane = tid & 31, half = lane >> 4, ln = lane & 15;
  int mrow0 = blockIdx.x * 16;
  int ncol = wid * 16 + ln;
  const float* Ap = A + (size_t)(mrow0 + ln) * lda + 2 * half;
  const float* Wp = W + (size_t)ncol * ldw + 2 * half;
  __builtin_prefetch(Ap, 0, 1);
  v8f acc = vzero8();
  for (int k0 = 0; k0 < Kdim; k0 += 4) {
    v2f a = *(const v2f*)(Ap + k0);
    v2f b = *(const v2f*)(Wp + k0);
    acc = wmma4(a, b, acc);
  }
  float bv = bias[ncol];
  if (mode == 0) {
    for (int r = 0; r < 8; r++)
      Out[(size_t)(mrow0 + r + 8 * half) * 128 + ncol] = acc[r] + bv;
  } else {
    int h = wid;
    for (int r = 0; r < 8; r++) {
      int row = mrow0 + r + 8 * half;
      int b_ = row >> 10, mi = row & (MM - 1);
      Out[(size_t)((b_ * HH + h) * MM + mi) * DD + ln] = acc[r] + bv;
    }
  }
}

// ---------------------------------------------------------------------------
// 8) BN1 finalize + relu
// ---------------------------------------------------------------------------
__global__ void bn1_final(const float* __restrict__ xm, const float* __restrict__ sum,
                          const float* __restrict__ sq, const float* __restrict__ g,
                          const float* __restrict__ be, float* __restrict__ out) {
  int i = blockIdx.x * 256 + threadIdx.x;
  int c = i & 127;
  const float invc = 1.f / 8192.f;
  float mu = sum[c] * invc;
  float var = sq[c] * invc - mu * mu;
  float v = (xm[i] - mu) * rsqrtf(var + 1e-5f) * g[c] + be[c];
  out[i] = fmaxf(v, 0.f);
}

// ---------------------------------------------------------------------------
// 9) Flash-style MHA over m=1024, d=16: wave per 16-row query tile.
// ---------------------------------------------------------------------------
__global__ void __launch_bounds__(256)
mha_flash(const float* __restrict__ qh, const float* __restrict__ kh,
          const float* __restrict__ vh, float* __restrict__ out) {
  int tid = threadIdx.x;
  int wid = tid >> 5, lane = tid & 31, half = lane >> 4, ln = lane & 15;
  int Wg = blockIdx.x * 8 + wid;     // 4096 wave tasks
  int b = Wg >> 9;
  int rem = Wg & 511;
  int h = rem >> 6;
  int qt = rem & 63;

  alignas(16) __shared__ float S[8][16][16];
  __shared__ float rowA[8][16];
  __shared__ float rowL[8][16];

  const float* qbase = qh + (size_t)((b * HH + h) * MM + qt * 16 + ln) * DD;
  v2f q2[4];
#pragma unroll
  for (int i = 0; i < 4; i++) q2[i] = *(const v2f*)(qbase + 4 * i + 2 * half);

  v8f o = vzero8();
  float mrun = -3e38f, lrun = 0.f;
  const float scale = 0.25f;  // d^-0.5

  for (int j = 0; j < 64; j++) {
    // S = (Q @ K^T) * scale
    const float* kb = kh + (size_t)((b * HH + h) * MM + j * 16 + ln) * DD;
    v8f s = vzero8();
#pragma unroll
    for (int i = 0; i < 4; i++) {
      v2f bf = *(const v2f*)(kb + 4 * i + 2 * half);
      s = wmma4(q2[i], bf, s);
    }
#pragma unroll
    for (int r = 0; r < 8; r++) S[wid][r + 8 * half][ln] = s[r] * scale;
    __syncthreads();

    if (lane < 16) {
      float mloc = -3e38f;
      for (int n = 0; n < 16; n++) mloc = fmaxf(mloc, S[wid][lane][n]);
      float mnew = fmaxf(mrun, mloc);
      float al = __expf(mrun - mnew);
      float ps = 0.f;
      for (int n = 0; n < 16; n++) {
        float p = __expf(S[wid][lane][n] - mnew);
        S[wid][lane][n] = p;
        ps += p;
      }
      lrun = lrun * al + ps;
      mrun = mnew;
      rowA[wid][lane] = al;
    }
    __syncthreads();

#pragma unroll
    for (int r = 0; r < 8; r++) o[r] *= rowA[wid][r + 8 * half];

    // O += P @ V
    const float* vb = vh + (size_t)((b * HH + h) * MM + j * 16) * DD;
#pragma unroll
    for (int i = 0; i < 4; i++) {
      v2f a = *(const v2f*)(&S[wid][ln][4 * i + 2 * half]);
      v2f bf;
      bf.x = vb[(4 * i + 2 * half) * DD + ln];
      bf.y = vb[(4 * i + 2 * half + 1) * DD + ln];
      o = wmma4(a, bf, o);
    }
    __syncthreads();
  }

  if (lane < 16) rowL[wid][lane] = lrun;
  __syncthreads();
#pragma unroll
  for (int r = 0; r < 8; r++) {
    float l = rowL[wid][r + 8 * half];
    int mrow = qt * 16 + r + 8 * half;
    out[(size_t)(b * MM + mrow) * CO + h * DD + ln] = o[r] / l;
  }
}

// ---------------------------------------------------------------------------
// 10) 3-NN interpolation back to N points -> d_out (B, 128, N)
// ---------------------------------------------------------------------------
__global__ void nn_interp(const float* __restrict__ coords, const float* __restrict__ cc,
                          const float* __restrict__ xcf, float* __restrict__ out) {
  int b = blockIdx.x >> 5;                         // 32 blocks per batch
  int n = ((blockIdx.x & 31) << 8) + threadIdx.x;  // 0..8191
  __shared__ float cx[MM], cy[MM], cz[MM];
  for (int t = threadIdx.x; t < MM; t += 256) {
    cx[t] = cc[(b * 3 + 0) * MM + t];
    cy[t] = cc[(b * 3 + 1) * MM + t];
    cz[t] = cc[(b * 3 + 2) * MM + t];
  }
  __syncthreads();
  float px = coords[(size_t)(b * 3 + 0) * NN + n];
  float py = coords[(size_t)(b * 3 + 1) * NN + n];
  float pz = coords[(size_t)(b * 3 + 2) * NN + n];
  float d0 = 3e38f, d1 = 3e38f, d2 = 3e38f;
  int i0 = 0, i1 = 0, i2 = 0;
  for (int m = 0; m < MM; m++) {
    float dx = px - cx[m], dy = py - cy[m], dz = pz - cz[m];
    float d = dx * dx + dy * dy + dz * dz;
    if (d < d0)      { d2 = d1; i2 = i1; d1 = d0; i1 = i0; d0 = d; i0 = m; }
    else if (d < d1) { d2 = d1; i2 = i1; d1 = d; i1 = m; }
    else if (d < d2) { d2 = d; i2 = m; }
  }
  float w0 = 1.f / (d0 + 1e-8f), w1 = 1.f / (d1 + 1e-8f), w2 = 1.f / (d2 + 1e-8f);
  float ws = w0 + w1 + w2;
  w0 /= ws; w1 /= ws; w2 /= ws;
  const float* f0 = xcf + (size_t)(b * MM + i0) * CO;
  const float* f1 = xcf + (size_t)(b * MM + i1) * CO;
  const float* f2 = xcf + (size_t)(b * MM + i2) * CO;
  for (int c = 0; c < CO; c++)
    out[(size_t)(b * CO + c) * NN + n] = w0 * f0[c] + w1 * f1[c] + w2 * f2[c];
}

// ---------------------------------------------------------------------------
// Host orchestration
// ---------------------------------------------------------------------------
extern "C" void kernel_launch(void* const* d_in, const int* in_sizes, int n_in,
                              void* d_out, int out_size, void* d_ws, size_t ws_size,
                              hipStream_t stream) {
  const float* feats  = (const float*)d_in[0];   // (8,64,8192)
  const float* coords = (const float*)d_in[1];   // (8,3,8192)
  const float* Wq  = (const float*)d_in[2];  const float* bq  = (const float*)d_in[3];
  const float* Wk  = (const float*)d_in[4];  const float* bk  = (const float*)d_in[5];
  const float* Wv  = (const float*)d_in[6];  const float* bv  = (const float*)d_in[7];
  const float* Wtc = (const float*)d_in[8];  const float* btc = (const float*)d_in[9];
  const float* bn2g = (const float*)d_in[10]; const float* bn2b = (const float*)d_in[11];
  const float* Wmid = (const float*)d_in[12]; const float* bmid = (const float*)d_in[13];
  const float* bn1g = (const float*)d_in[14]; const float* bn1b = (const float*)d_in[15];
  const float* Wqm = (const float*)d_in[16]; const float* bqm = (const float*)d_in[17];
  const float* Wkm = (const float*)d_in[18]; const float* bkm = (const float*)d_in[19];
  const float* Wvm = (const float*)d_in[20]; const float* bvm = (const float*)d_in[21];
  const float* Wp  = (const float*)d_in[22]; const float* bp  = (const float*)d_in[23];
  float* out = (float*)d_out;

  char* base = (char*)d_ws;
  size_t off = 0;
  auto alloc = [&](size_t bytes) -> void* {
    void* p = base + off;
    off = (off + bytes + 255) & ~(size_t)255;
    return p;
  };
  int*   fpsIdx = (int*)  alloc((size_t)BB * MM * 4);
  float* cc     = (float*)alloc((size_t)BB * 3 * MM * 4);
  float* cfR    = (float*)alloc((size_t)BB * MM * CI * 4);
  int*   nbr    = (int*)  alloc((size_t)BB * MM * KK * 4);
  float* xr     = (float*)alloc((size_t)BB * MM * CO * KK * 4);   // 128 MB
  float* psum   = (float*)alloc((size_t)BB * MM * CO * 4);
  float* psq    = (float*)alloc((size_t)BB * MM * CO * 4);
  float* bn2s   = (float*)alloc(CO * 4);
  float* bn2q   = (float*)alloc(CO * 4);
  float* xmIn   = (float*)alloc((size_t)BB * MM * 256 * 4);
  float* xmB    = (float*)alloc((size_t)BB * MM * CO * 4);
  float* bn1s   = (float*)alloc(CO * 4);
  float* bn1q   = (float*)alloc(CO * 4);
  float* xr2    = (float*)alloc((size_t)BB * MM * CO * 4);
  float* qhB    = (float*)alloc((size_t)BB * MM * CO * 4);
  float* khB    = (float*)alloc((size_t)BB * MM * CO * 4);
  float* vhB    = (float*)alloc((size_t)BB * MM * CO * 4);
  float* aOut   = (float*)alloc((size_t)BB * MM * CO * 4);
  float* xcf    = (float*)alloc((size_t)BB * MM * CO * 4);
  (void)ws_size; (void)in_sizes; (void)n_in; (void)out_size;

  // 1) FPS
  fps_kernel<<<BB, 256, 0, stream>>>(coords, fpsIdx);
  // 2) gather centers
  gather_centers<<<(BB * MM) / 256, 256, 0, stream>>>(coords, feats, fpsIdx, cc, cfR);
  // 3) ball query
  ball_query<<<(BB * MM) / 256, 256, 0, stream>>>(coords, cc, nbr);
  // 4) fused local attention (q/k/v + attn + Wtc) + BN2 partials
  local_attn<<<BB * MM, 256, 0, stream>>>(feats, cfR, nbr, Wq, bq, Wk, bk, Wv, bv,
                                          Wtc, btc, xr, psum, psq);
  // 5) BN2 stats (deterministic)
  col_reduce<<<CO, 256, 0, stream>>>(psum, BB * MM, bn2s, bn2q, 0);
  col_reduce<<<CO, 256, 0, stream>>>(psq,  BB * MM, bn2q, bn2q, 0);
  // 6) BN2 finalize + relu + max/mean pool
  bn2_pool<<<BB * MM, CO, 0, stream>>>(xr, bn2s, bn2q, bn2g, bn2b, xmIn);
  // 7) Wmid GEMM (K=256)
  gemm_wmma<<<(BB * MM) / 16, 256, 0, stream>>>(xmIn, 256, Wmid, 256, bmid, xmB, 256, 0);
  // 8) BN1 stats + finalize
  col_reduce<<<CO, 256, 0, stream>>>(xmB, BB * MM, bn1s, bn1q, 1);
  bn1_final<<<(BB * MM * CO) / 256, 256, 0, stream>>>(xmB, bn1s, bn1q, bn1g, bn1b, xr2);
  // 9) MHA projections (head-split stores)
  gemm_wmma<<<(BB * MM) / 16, 256, 0, stream>>>(cfR, CI,  Wqm, CI,  bqm, qhB, CI,  1);
  gemm_wmma<<<(BB * MM) / 16, 256, 0, stream>>>(xr2, CO,  Wkm, CO,  bkm, khB, CO, 1);
  gemm_wmma<<<(BB * MM) / 16, 256, 0, stream>>>(xr2, CO,  Wvm, CO,  bvm, vhB, CO, 1);
  // 10) flash attention (4096 wave tasks / 8 waves per block)
  mha_flash<<<512, 256, 0, stream>>>(qhB, khB, vhB, aOut);
  // 11) Wp projection
  gemm_wmma<<<(BB * MM) / 16, 256, 0, stream>>>(aOut, CO, Wp, CO, bp, xcf, CO, 0);
  // 12) 3-NN interpolation -> output (B,128,N)
  nn_interp<<<BB * 32, 256, 0, stream>>>(coords, cc, xcf, out);
}